// ASGSCriterion_62715112456404
// MI455X (gfx1250) — compile-verified
//
#include <hip/hip_runtime.h>
#include <hip/hip_bf16.h>
#include <math.h>

#define B_   64
#define Q_   1000
#define D_   512
#define C_   81
#define NK_  80
#define KB_  5
#define MK_  5
#define DELTA_ 0.6f
#define TAU_   0.1f
#define NEGF  (-3.402823466e38f)
#define ATS  516   // padded LDS row stride (floats): 516%64=4 -> conflict-free frags, 16B aligned

typedef float v2f __attribute__((ext_vector_type(2)));
typedef float v8f __attribute__((ext_vector_type(8)));
typedef unsigned int v4u __attribute__((ext_vector_type(4)));
typedef int v4i __attribute__((ext_vector_type(4)));
typedef int v8i __attribute__((ext_vector_type(8)));

__device__ __forceinline__ float wred(float v){
#pragma unroll
  for (int o = 16; o > 0; o >>= 1) v += __shfl_xor(v, o, 32);
  return v;
}
__device__ __forceinline__ float wmaxr(float v){
#pragma unroll
  for (int o = 16; o > 0; o >>= 1) v = fmaxf(v, __shfl_xor(v, o, 32));
  return v;
}

template<int N>
__device__ __forceinline__ void topk_ins(float (&vals)[N], int (&idxs)[N], float v, int id){
  if (!(v > vals[N-1])) return;      // strict > : stable ties (earlier index wins), matches jax top_k
  int p = N-1;
  while (p > 0 && v > vals[p-1]) { vals[p] = vals[p-1]; idxs[p] = idxs[p-1]; --p; }
  vals[p] = v; idxs[p] = id;
}

// ---- Tensor Data Mover: DMA a 16x512 f32 row-tile into LDS with +2dword pad per 256 dwords
// (LDS row stride becomes 2064B = 516 floats; one +8B shift after element 255 within a row).
__device__ __forceinline__ void tdm_load_tile(const float* gaddr, unsigned lds_byte_off,
                                              int rows_avail){
  unsigned long long ga = (unsigned long long)(size_t)gaddr;
  v4u g0; v8i g1; v4i g2 = {0,0,0,0}; v4i g3 = {0,0,0,0};
  v8i g4 = {0,0,0,0,0,0,0,0};                              // unused VADDR4 slot (NULL)
  g0[0] = 1u;                                              // count=1 (valid user descriptor)
  g0[1] = lds_byte_off;                                    // LDS destination (bytes)
  g0[2] = (unsigned)(ga & 0xFFFFFFFFull);                  // global_addr[31:0]
  g0[3] = (unsigned)((ga >> 32) & 0x1FFFFFFull) | (2u<<30);// global_addr[56:32] | type=2
  g1[0] = (int)((2u << 16)    // data_size = 4 bytes
              | (1u << 20)    // pad_enable
              | (7u << 22)    // pad_interval: 256 DWORDs
              | (1u << 25));  // pad_amount:   2 DWORDs
  g1[1] = (int)(512u << 16);                               // tensor_dim0 = 512 (lo16)
  g1[2] = (int)(((unsigned)rows_avail & 0xFFFFu) << 16);   // dim0 hi16=0 | tensor_dim1 lo16
  g1[3] = (int)(512u << 16);                               // dim1 hi16=0 | tile_dim0 = 512
  g1[4] = 16;                                              // tile_dim1 = 16, tile_dim2 = 0
  g1[5] = 512;                                             // tensor_dim0_stride lo32
  g1[6] = 0;
  g1[7] = 0;
  __builtin_amdgcn_tensor_load_to_lds(g0, g1, g2, g3, g4, 0);
}

// ---------------- init: zero accumulators ----------------
__global__ __launch_bounds__(256)
void k_init(float* col_sum, float* pos_sum, float* accum){
  int t = threadIdx.x;
  if (t < C_) { col_sum[t] = 0.f; pos_sum[t] = 0.f; }
  if (t < 4)  accum[t] = 0.f;
}

// ---------------- row norms of obj_embs ----------------
__global__ __launch_bounds__(256)
void k_norm(const float* __restrict__ emb, float* __restrict__ invn, float* __restrict__ e2){
  int w = threadIdx.x >> 5, lane = threadIdx.x & 31;
  int row = blockIdx.x * 8 + w;
  if (row >= B_ * Q_) return;
  const float* er = emb + (size_t)row * D_;
  float s = 0.f;
#pragma unroll 4
  for (int t = lane; t < D_; t += 32) { float v = er[t]; s += v * v; }
  s = wred(s);
  if (lane == 0) { e2[row] = s; invn[row] = 1.f / fmaxf(sqrtf(s), 1e-12f); }
}

// ---------------- prototype norms ----------------
__global__ __launch_bounds__(32)
void k_proto(const float* __restrict__ protos, float* __restrict__ proto_n, float* __restrict__ p2){
  int c = blockIdx.x, lane = threadIdx.x;
  const float* pr = protos + (size_t)c * D_;
  float s = 0.f;
#pragma unroll 4
  for (int t = lane; t < D_; t += 32) { float v = pr[t]; s += v * v; }
  s = wred(s);
  float iv = 1.f / fmaxf(sqrtf(s), 1e-12f);
  for (int t = lane; t < D_; t += 32) proto_n[(size_t)c * D_ + t] = pr[t] * iv;
  if (c < NK_ && lane == 0) p2[c] = s;
}

// ---------------- boundary pick: masked top-5 distances per (b,k) ----------------
__global__ __launch_bounds__(32)
void k_boundary(const float* __restrict__ emb, const float* __restrict__ protos,
                const int* __restrict__ labels, const float* __restrict__ e2,
                const float* __restrict__ p2, int* __restrict__ bidx, int* __restrict__ bvalid){
  int pair = blockIdx.x;               // b*NK + k
  int b = pair / NK_, k = pair % NK_;
  int lane = threadIdx.x;
  const float* prow = protos + (size_t)k * D_;
  const int* lab = labels + b * Q_;
  float d5[KB_]; int i5[KB_]; int cnt = 0;
#pragma unroll
  for (int j = 0; j < KB_; ++j) { d5[j] = NEGF; i5[j] = 0; }
  float pk2 = p2[k];
  for (int q0 = 0; q0 < Q_; q0 += 32) {
    int q = q0 + lane;
    int lv = (q < Q_) ? lab[q] : -1;
    unsigned long long m = __ballot(lv == k);
    while (m) {
      int j = __ffsll(m) - 1; m &= m - 1;
      int qq = q0 + j;
      const float* er = emb + ((size_t)b * Q_ + qq) * D_;
      float s = 0.f;
#pragma unroll 4
      for (int t = lane; t < D_; t += 32) s += er[t] * prow[t];
      s = wred(s);                              // broadcast to all lanes
      float dist = sqrtf(fmaxf(e2[b * Q_ + qq] - 2.f * s + pk2, 1e-12f));
      ++cnt;
      topk_ins<KB_>(d5, i5, dist, qq);          // all lanes hold identical state
    }
  }
  int base = pair * KB_;
  if (lane < KB_) {
    bidx[base + lane]   = (lane < (cnt < KB_ ? cnt : KB_)) ? i5[lane] : 0;
    bvalid[base + lane] = (lane < cnt) ? 1 : 0;
  }
}

// ---------------- fused WMMA sim GEMM + masked top-5 kNN, TDM double-buffered B ----------------
__global__ __launch_bounds__(32)
void k_simknn(const float* __restrict__ emb, const float* __restrict__ invn,
              const int* __restrict__ labels, const int* __restrict__ bidx,
              int* __restrict__ nb_i, float* __restrict__ nb_sim){
  extern __shared__ float smem[];
  float* As  = smem;                       // 16*ATS : normalized boundary rows
  float* Bs0 = smem + 16 * ATS;            // 16*ATS : TDM buffer 0
  float* Bs1 = Bs0  + 16 * ATS;            // 16*ATS : TDM buffer 1
  float* Ct  = Bs1  + 16 * ATS;            // 16*17  : C tile
  int*   Ls  = (int*)(Ct + 16 * 17);       // Q_     : labels

  int b    = blockIdx.y;
  int rt   = blockIdx.x;                   // row-tile 0..24 (400/16)
  int lane = threadIdx.x;

  for (int q = lane; q < Q_; q += 32) Ls[q] = labels[b * Q_ + q];

  // stage A: rows rt*16+m = gathered, l2-normalized boundary embeddings
  for (int m = 0; m < 16; ++m) {
    int rrow = rt * 16 + m;                         // k*5+i
    int qi = bidx[b * NK_ * KB_ + rrow];
    float sc = invn[b * Q_ + qi];
    const float4* src = (const float4*)(emb + ((size_t)b * Q_ + qi) * D_);
    float4* dst = (float4*)(As + m * ATS);
    for (int t = lane; t < D_ / 4; t += 32) {
      float4 v = src[t];
      v.x *= sc; v.y *= sc; v.z *= sc; v.w *= sc;
      dst[t] = v;
    }
  }

  const float* batchbase = emb + (size_t)b * Q_ * D_;
  // kick off chunk 0 DMA into Bs0
  tdm_load_tile(batchbase, (unsigned)(size_t)(void*)Bs0, Q_);
  __syncthreads();

  int m  = lane & 15;
  int kh = (lane >> 4) << 1;               // 0 or 2 (K sub-offset per fragment)
  const float* arow = As + m * ATS + kh;

  float s5[MK_]; int q5[MK_];
#pragma unroll
  for (int j = 0; j < MK_; ++j) { s5[j] = NEGF; q5[j] = 0; }

  const int NCH = (Q_ + 15) / 16;          // 63
  int cur = 0;
  for (int ci = 0; ci < NCH; ++ci) {
    int q0 = ci * 16;
    float* bcur  = cur ? Bs1 : Bs0;
    float* bnext = cur ? Bs0 : Bs1;
    if (ci + 1 < NCH) {
      // prefetch next tile via TDM; tensor ops complete in-order per wave,
      // so tensorcnt<=1 guarantees the *current* buffer is ready.
      tdm_load_tile(batchbase + (size_t)(q0 + 16) * D_,
                    (unsigned)(size_t)(void*)bnext, Q_ - (q0 + 16));
      __builtin_amdgcn_s_wait_tensorcnt(1);
    } else {
      __builtin_amdgcn_s_wait_tensorcnt(0);
    }

    int qn = q0 + (lane & 15);
    int qc = qn < Q_ ? qn : Q_ - 1;
    float scB = invn[b * Q_ + qc];          // rows beyond Q_ are DMA-zeroed anyway
    const float* brow = bcur + (lane & 15) * ATS + kh;

    v8f acc = {};
#pragma unroll 4
    for (int kk = 0; kk < 256; kk += 4) {   // first half-row (before the +8B pad shift)
      v2f a, bb;
      a[0]  = arow[kk];        a[1]  = arow[kk + 1];
      bb[0] = brow[kk] * scB;  bb[1] = brow[kk + 1] * scB;
      acc = __builtin_amdgcn_wmma_f32_16x16x4_f32(false, a, false, bb,
                                                  (short)0, acc, false, false);
    }
    const float* brow2 = brow + 2;          // skip mid-row pad (2 DWORDs)
#pragma unroll 4
    for (int kk = 256; kk < 512; kk += 4) {
      v2f a, bb;
      a[0]  = arow[kk];         a[1]  = arow[kk + 1];
      bb[0] = brow2[kk] * scB;  bb[1] = brow2[kk + 1] * scB;
      acc = __builtin_amdgcn_wmma_f32_16x16x4_f32(false, a, false, bb,
                                                  (short)0, acc, false, false);
    }

    // C layout: VGPR j -> row j (lanes 0-15) / row j+8 (lanes 16-31), col = lane%16
    int rofs = (lane >> 4) * 8;
#pragma unroll
    for (int j = 0; j < 8; ++j) Ct[(j + rofs) * 17 + (lane & 15)] = acc[j];
    __syncthreads();

    if (lane < 16) {                 // lane owns row m=lane; masked top-5 over this chunk
#pragma unroll
      for (int n = 0; n < 16; ++n) {
        int q = q0 + n;
        if (q < Q_ && Ls[q] >= NK_) topk_ins<MK_>(s5, q5, Ct[lane * 17 + n], q);
      }
    }
    __syncthreads();
    cur ^= 1;
  }

  if (lane < 16) {
    int rrow = rt * 16 + lane;
    int base = (b * NK_ * KB_ + rrow) * MK_;
#pragma unroll
    for (int j = 0; j < MK_; ++j) { nb_i[base + j] = q5[j]; nb_sim[base + j] = s5[j]; }
  }
}

// ---------------- g_bar + logits + CE (loss_sul accumulation) ----------------
__global__ __launch_bounds__(32)
void k_sul(const float* __restrict__ emb, const float* __restrict__ cls_w,
           const float* __restrict__ cls_b, const int* __restrict__ bidx,
           const int* __restrict__ bvalid, const int* __restrict__ nb_i,
           const float* __restrict__ nb_sim, float* __restrict__ accum){
  int r = blockIdx.x;                 // 0 .. B*400-1
  int b = r / (NK_ * KB_);
  int lane = threadIdx.x;
  int qb = bidx[r];

  float g[16];
  const float* brow = emb + ((size_t)b * Q_ + qb) * D_;
#pragma unroll
  for (int t = 0; t < 16; ++t) g[t] = brow[lane + 32 * t];

  int cnt = 0;
#pragma unroll
  for (int j = 0; j < MK_; ++j) {
    if (nb_sim[r * MK_ + j] > DELTA_) {
      ++cnt;
      const float* nrow = emb + ((size_t)b * Q_ + nb_i[r * MK_ + j]) * D_;
#pragma unroll
      for (int t = 0; t < 16; ++t) g[t] += nrow[lane + 32 * t];
    }
  }
  float invc = 1.f / (1.f + (float)cnt);
#pragma unroll
  for (int t = 0; t < 16; ++t) g[t] *= invc;

  float lmy[3] = {NEGF, NEGF, NEGF};
  float l80 = 0.f;
  for (int c = 0; c < C_; ++c) {
    const float* w = cls_w + (size_t)c * D_;
    float s = 0.f;
#pragma unroll
    for (int t = 0; t < 16; ++t) s += g[t] * w[lane + 32 * t];
    s = wred(s) + cls_b[c];
    if ((c & 31) == lane) lmy[c >> 5] = s;
    if (c == C_ - 1) l80 = s;
  }
  float M = fmaxf(fmaxf(lmy[0], lmy[1]), lmy[2]);
  M = wmaxr(M);
  float se = expf(lmy[0] - M) + expf(lmy[1] - M) + expf(lmy[2] - M); // invalid -> exp(-huge)=0
  se = wred(se);
  float ce = M + logf(se) - l80;      // -log_softmax[...,UNKNOWN]

  if (bvalid[r] && cnt > 0 && lane == 0) {
    atomicAdd(&accum[0], ce);
    atomicAdd(&accum[1], 1.f);
  }
}

// ---------------- p_neg from prototype similarity matrix ----------------
__global__ __launch_bounds__(32)
void k_pneg(const float* __restrict__ proto_n, float* __restrict__ p_neg){
  int c = blockIdx.x, lane = threadIdx.x;
  const float* pc = proto_n + (size_t)c * D_;
  float acc = 0.f, diag = 0.f;
  for (int c2 = 0; c2 < C_; ++c2) {
    const float* pr = proto_n + (size_t)c2 * D_;
    float s = 0.f;
#pragma unroll 4
    for (int t = lane; t < D_; t += 32) s += pc[t] * pr[t];
    s = wred(s);
    float e = expf(s / TAU_);
    acc += e; if (c2 == c) diag = e;
  }
  if (lane == 0) p_neg[c] = acc - diag;
}

// ---------------- masked S-pass: col_sum / pos_sum / pos_exp ----------------
__global__ __launch_bounds__(256)
void k_spass(const float* __restrict__ emb, const float* __restrict__ invn,
             const int* __restrict__ labels, const float* __restrict__ proto_n,
             float* __restrict__ col_sum, float* __restrict__ pos_sum,
             float* __restrict__ pos_exp){
  __shared__ float cs[C_], ps[C_];
  int tid = threadIdx.x;
  for (int c = tid; c < C_; c += 256) { cs[c] = 0.f; ps[c] = 0.f; }
  __syncthreads();
  int w = tid >> 5, lane = tid & 31;
  int row = blockIdx.x * 8 + w;
  if (row < B_ * Q_) {
    int lab = labels[row];
    if (lab < NK_) {
      const float* er = emb + (size_t)row * D_;
      float sc = invn[row];
      for (int c = 0; c < C_; ++c) {
        const float* pr = proto_n + (size_t)c * D_;
        float s = 0.f;
#pragma unroll 4
        for (int t = lane; t < D_; t += 32) s += er[t] * pr[t];
        s = wred(s) * sc * (1.f / TAU_);
        float e = expf(s);
        if (lane == 0) {
          atomicAdd(&cs[c], e);
          if (c == lab) { atomicAdd(&ps[c], e); pos_exp[row] = e; }
        }
      }
    }
  }
  __syncthreads();
  for (int c = tid; c < C_; c += 256) {
    if (cs[c] != 0.f) atomicAdd(&col_sum[c], cs[c]);
    if (ps[c] != 0.f) atomicAdd(&pos_sum[c], ps[c]);
  }
}

// ---------------- E = p_neg + col_sum - pos_sum ----------------
__global__ __launch_bounds__(128)
void k_E(const float* p_neg, const float* col_sum, const float* pos_sum, float* E){
  int c = threadIdx.x;
  if (c < C_) E[c] = p_neg[c] + col_sum[c] - pos_sum[c];
}

// ---------------- per-row CEC loss accumulation ----------------
__global__ __launch_bounds__(256)
void k_cec(const int* __restrict__ labels, const float* __restrict__ pos_exp,
           const float* __restrict__ E, float* __restrict__ accum){
  __shared__ float ls, lc;
  if (threadIdx.x == 0) { ls = 0.f; lc = 0.f; }
  __syncthreads();
  int i = blockIdx.x * 256 + threadIdx.x;
  if (i < B_ * Q_) {
    int lab = labels[i];
    if (lab < NK_) {
      float pe = pos_exp[i];
      float ratio = pe / (pe + E[lab] + 1e-8f);
      atomicAdd(&ls, -logf(ratio));
      atomicAdd(&lc, 1.f);
    }
  }
  __syncthreads();
  if (threadIdx.x == 0) { atomicAdd(&accum[2], ls); atomicAdd(&accum[3], lc); }
}

// ---------------- finalize ----------------
__global__ void k_fin(const float* accum, float* out){
  if (threadIdx.x == 0) {
    out[0] = (accum[1] > 0.f) ? accum[0] / fmaxf(accum[1], 1.f) : 0.f;
    out[1] = (accum[3] > 0.f) ? accum[2] / fmaxf(accum[3], 1.f) : 0.f;
  }
}

extern "C" void kernel_launch(void* const* d_in, const int* in_sizes, int n_in,
                              void* d_out, int out_size, void* d_ws, size_t ws_size,
                              hipStream_t stream) {
  const float* emb     = (const float*)d_in[0];   // (B,Q,D)
  const float* protos  = (const float*)d_in[1];   // (C,D)
  const float* cls_w   = (const float*)d_in[2];   // (C,D)
  const float* cls_b   = (const float*)d_in[3];   // (C,)
  const int*   labels  = (const int*)  d_in[4];   // (B,Q)
  float* out = (float*)d_out;

  float* w = (float*)d_ws;
  float* invn    = w;  w += B_ * Q_;
  float* e2      = w;  w += B_ * Q_;
  float* proto_n = w;  w += C_ * D_;
  float* p2      = w;  w += NK_;
  int*   bidx    = (int*)w; w += B_ * NK_ * KB_;
  int*   bvalid  = (int*)w; w += B_ * NK_ * KB_;
  int*   nb_i    = (int*)w; w += B_ * NK_ * KB_ * MK_;
  float* nb_sim  = w;  w += B_ * NK_ * KB_ * MK_;
  float* p_neg   = w;  w += C_;
  float* col_sum = w;  w += C_;
  float* pos_sum = w;  w += C_;
  float* E       = w;  w += C_;
  float* pos_exp = w;  w += B_ * Q_;
  float* accum   = w;  w += 4;

  size_t smem_knn = (size_t)(16 * ATS * 3 + 16 * 17) * sizeof(float) + (size_t)Q_ * sizeof(int);

  k_init    <<<1, 256, 0, stream>>>(col_sum, pos_sum, accum);
  k_norm    <<<(B_ * Q_ + 7) / 8, 256, 0, stream>>>(emb, invn, e2);
  k_proto   <<<C_, 32, 0, stream>>>(protos, proto_n, p2);
  k_boundary<<<B_ * NK_, 32, 0, stream>>>(emb, protos, labels, e2, p2, bidx, bvalid);
  k_simknn  <<<dim3((NK_ * KB_) / 16, B_), 32, smem_knn, stream>>>(emb, invn, labels, bidx, nb_i, nb_sim);
  k_sul     <<<B_ * NK_ * KB_, 32, 0, stream>>>(emb, cls_w, cls_b, bidx, bvalid, nb_i, nb_sim, accum);
  k_pneg    <<<C_, 32, 0, stream>>>(proto_n, p_neg);
  k_spass   <<<(B_ * Q_ + 7) / 8, 256, 0, stream>>>(emb, invn, labels, proto_n, col_sum, pos_sum, pos_exp);
  k_E       <<<1, 128, 0, stream>>>(p_neg, col_sum, pos_sum, E);
  k_cec     <<<(B_ * Q_ + 255) / 256, 256, 0, stream>>>(labels, pos_exp, E, accum);
  k_fin     <<<1, 32, 0, stream>>>(accum, out);
}